// OpenBoundary_18339510354342
// MI455X (gfx1250) — compile-verified
//
#include <hip/hip_runtime.h>

#define NPART 8192
#define MAXN  128
#define CUT2  0.015625f   // 0.125^2, exact in fp32
#define NWAVE 8           // waves per workgroup
#define COLS_PER_WAVE (NPART / NWAVE)       // 1024
#define TILES_PER_WAVE (COLS_PER_WAVE / 16) // 64

typedef __attribute__((ext_vector_type(2))) float v2f;
typedef __attribute__((ext_vector_type(8))) float v8f;

// wave32 cross-lane moves via LDS-permute unit (no LDS memory touched)
__device__ __forceinline__ float bcastf(float v, int srcLane) {
  return __int_as_float(__builtin_amdgcn_ds_bpermute(srcLane << 2, __float_as_int(v)));
}
__device__ __forceinline__ int lanei(int v, int srcLane) {
  return __builtin_amdgcn_ds_bpermute(srcLane << 2, v);
}

// One 16x16 tile of squared distances via V_WMMA_F32_16X16X4_F32.
// A row  = [-2x,-2y,-2z,1]   (lanes 0-15: K0/K1, lanes 16-31: K2/K3)
// B col  = [x,y,z,|p|^2]     (same half-wave split)
// C      = |p_row|^2  ->  D[i][j] = |pi|^2 - 2 pi.pj + |pj|^2 = d2
__device__ __forceinline__ v8f tile_d2(const float* __restrict__ P,
                                       v2f afrag, const float* cinit,
                                       int col0, int lane16, bool hi) {
  const int n = col0 + lane16;
  const float nx = P[3*n+0], ny = P[3*n+1], nz = P[3*n+2];
  const float nsq = nx*nx + ny*ny + nz*nz;
  v2f bfrag;
  bfrag.x = hi ? nz  : nx;
  bfrag.y = hi ? nsq : ny;
  v8f c;
#pragma unroll
  for (int r = 0; r < 8; ++r) c[r] = cinit[r];
  return __builtin_amdgcn_wmma_f32_16x16x4_f32(false, afrag, false, bfrag,
                                               (short)0, c, false, false);
}

// Zero the cell_indices block (all-zero for open boundary) and the max cell.
__global__ __launch_bounds__(256) void OpenBoundary_zero_kernel(int* __restrict__ out) {
  const long long i = (long long)blockIdx.x * blockDim.x + threadIdx.x;
  int4* cells = (int4*)(out + (long long)NPART * MAXN);   // 16B-aligned (4MB offset)
  const long long n4 = (long long)NPART * MAXN * 3 / 4;   // 786432
  if (i < n4) cells[i] = make_int4(0, 0, 0, 0);
  if (i == 0) out[(long long)NPART * MAXN * 4] = 0;       // actual_max_neighbours
}

__global__ __launch_bounds__(256) void OpenBoundary_nbr_kernel(const float* __restrict__ P,
                                                               int* __restrict__ out) {
  __shared__ int s_counts[NWAVE][16];
  __shared__ int s_prefix[NWAVE][16];

  const int lane   = threadIdx.x & 31;
  const int wave   = threadIdx.x >> 5;
  const int lane16 = lane & 15;
  const bool hi    = lane >= 16;
  const int  hi8   = hi ? 8 : 0;

  const int rowBase = blockIdx.x * 16;

  // ---- A fragment for this row block (shared by all waves in the WG) ----
  const int m = rowBase + lane16;
  const float mx = P[3*m+0], my = P[3*m+1], mz = P[3*m+2];
  const float nsq_m = mx*mx + my*my + mz*mz;
  v2f afrag;
  afrag.x = hi ? (-2.0f * mz) : (-2.0f * mx);
  afrag.y = hi ? 1.0f         : (-2.0f * my);

  // C init: |p_row|^2 for each of my 8 C-registers (rows r / r+8 per half-wave)
  float cinit[8];
#pragma unroll
  for (int r = 0; r < 8; ++r) cinit[r] = bcastf(nsq_m, r + hi8);

  const int colStart = wave * COLS_PER_WAVE;

  // =========== pass 1: count matches per row (no ballots, no self test) ===========
  int cntreg[8];
#pragma unroll
  for (int r = 0; r < 8; ++r) cntreg[r] = 0;

  for (int t = 0; t < TILES_PER_WAVE; ++t) {
    const int col0 = colStart + t * 16;
    const v8f c = tile_d2(P, afrag, cinit, col0, lane16, hi);
#pragma unroll
    for (int r = 0; r < 8; ++r) cntreg[r] += (c[r] <= CUT2) ? 1 : 0;
  }

  // butterfly-reduce each counter across the 16 lanes of my half
#pragma unroll
  for (int mk = 1; mk < 16; mk <<= 1) {
#pragma unroll
    for (int r = 0; r < 8; ++r) cntreg[r] += lanei(cntreg[r], lane ^ mk);
  }
  if (lane16 == 0) {
#pragma unroll
    for (int r = 0; r < 8; ++r) {
      // subtract the self pair (always within cutoff) if its column is in my stripe
      const int rowG = rowBase + r + hi8;
      const int selfIn = ((unsigned)(rowG - colStart) < (unsigned)COLS_PER_WAVE) ? 1 : 0;
      s_counts[wave][r + hi8] = cntreg[r] - selfIn;
    }
  }
  __syncthreads();

  // ---- exclusive prefix across waves; padding + global max (wave 0) ----
  if (wave == 0 && lane < 16) {
    const int rowG = rowBase + lane;
    int tot = 0;
#pragma unroll
    for (int w = 0; w < NWAVE; ++w) { s_prefix[w][lane] = tot; tot += s_counts[w][lane]; }
    const int fillStart = tot < MAXN ? tot : MAXN;
    for (int p = fillStart; p < MAXN; ++p) out[rowG * MAXN + p] = -1;
    atomicMax(&out[(long long)NPART * MAXN * 4], tot);
  }
  __syncthreads();

  // per-lane replicated running offsets for my 8 rows (broadcast LDS read)
  int offreg[8];
#pragma unroll
  for (int r = 0; r < 8; ++r) offreg[r] = s_prefix[wave][r + hi8];

  // =========== pass 2: recompute tiles and scatter compacted indices ===========
  for (int t = 0; t < TILES_PER_WAVE; ++t) {
    const int col0 = colStart + t * 16;
    const v8f c = tile_d2(P, afrag, cinit, col0, lane16, hi);
    const int j = col0 + lane16;
#pragma unroll
    for (int r = 0; r < 8; ++r) {
      const int rowM = rowBase + r + hi8;
      const bool pred = (c[r] <= CUT2) && (j != rowM);
      const unsigned bal = __builtin_amdgcn_ballot_w32(pred);
      const unsigned half = hi ? (bal >> 16) : (bal & 0xffffu);
      const int rank = __builtin_popcount(half & ((1u << lane16) - 1u));
      const int pos = offreg[r] + rank;
      if (pred && pos < MAXN) out[rowM * MAXN + pos] = j;
      offreg[r] += __builtin_popcount(half);   // advance my row's offset
    }
  }
}

extern "C" void kernel_launch(void* const* d_in, const int* in_sizes, int n_in,
                              void* d_out, int out_size, void* d_ws, size_t ws_size,
                              hipStream_t stream) {
  (void)in_sizes; (void)n_in; (void)d_ws; (void)ws_size; (void)out_size;
  const float* positions = (const float*)d_in[0];  // [8192,3] f32
  int* out = (int*)d_out;  // to_idx [8192,128] | cell_indices [8192,128,3] | max (1)

  // 786432 int4 stores to clear cell table + max cell
  OpenBoundary_zero_kernel<<<3072, 256, 0, stream>>>(out);
  // 512 row-blocks x 8 waves; WMMA f32 16x16x4 distance tiles
  OpenBoundary_nbr_kernel<<<512, 256, 0, stream>>>(positions, out);
}